// Model_90623809946045
// MI455X (gfx1250) — compile-verified
//
#include <hip/hip_runtime.h>
#include <math.h>

// Problem constants (match reference)
#define B_SZ   2048
#define NBUG   2000
#define DBUG   30
#define K_OUT  20
#define L_CL   40
#define P_SUB  2
#define NPAD   48   // L padded to 3 WMMA tiles of 16

typedef __attribute__((ext_vector_type(2))) float v2f;
typedef __attribute__((ext_vector_type(8))) float v8f;

__device__ __forceinline__ float sigmoidf_(float x) { return 1.0f / (1.0f + __expf(-x)); }
__device__ __forceinline__ float eluf_(float x)     { return x > 0.0f ? x : (__expf(x) - 1.0f); }

// ---------------------------------------------------------------------------
// Stage 1: w_act[k][n] = sigmoid(r[n] - ||microbe_locs[k] - mu_bug[n]||^2)
// Written in B-fragment-ready pairing: wactB[k/2][n] = {w[k][n], w[k+1][n]}
// (flat layout [NBUG/2][NPAD][2], cols 40..47 zeroed) so stage 2's B fragment
// is one contiguous b64 load per lane.
// ---------------------------------------------------------------------------
__global__ void wact_kernel(const float* __restrict__ ml,
                            const float* __restrict__ mu,
                            const float* __restrict__ r,
                            float* __restrict__ wactB) {
    int idx = blockIdx.x * blockDim.x + threadIdx.x;
    if (idx >= NBUG * NPAD) return;
    int k = idx / NPAD;   // bug index (GEMM K dim)
    int n = idx % NPAD;   // cluster column
    float w = 0.0f;
    if (n < L_CL) {
        float s = 0.0f;
#pragma unroll
        for (int d = 0; d < DBUG; ++d) {
            float t = ml[k * DBUG + d] - mu[n * DBUG + d];
            s += t * t;
        }
        w = sigmoidf_(r[n] - s);
    }
    wactB[(size_t)(k >> 1) * (NPAD * 2) + n * 2 + (k & 1)] = w;
}

// ---------------------------------------------------------------------------
// Stage 2: g = x @ w_act via V_WMMA_F32_16X16X4_F32.
// One wave per 16x16 tile; chain K=2000 in steps of 4.
// A-frag (16x4 f32): lane l holds M = l%16, K = 2*(l/16)+{0,1}  -> b64 from x row
// B-frag (4x16 f32): lane l holds N = l%16, K = 2*(l/16)+{0,1}  -> b64 from wactB
// C (16x16 f32, 8 VGPRs): vgpr r: lanes 0-15 -> M=r, lanes 16-31 -> M=r+8, N=l%16.
// ---------------------------------------------------------------------------
__global__ void g_gemm_kernel(const float* __restrict__ x,
                              const float* __restrict__ wactB,
                              float* __restrict__ g) {
    const int bt   = blockIdx.x;        // 0..127 batch tile
    const int nt   = blockIdx.y;        // 0..2   column tile
    const int lane = threadIdx.x;       // 0..31
    const int m    = lane & 15;
    const int hk   = (lane >> 4) * 2;   // 0 or 2
    const int row  = bt * 16 + m;
    const int n    = nt * 16 + m;

    const float2* wB = (const float2*)wactB;

    v8f c = {};
    for (int k0 = 0; k0 < NBUG; k0 += 4) {
        v2f a, b;
        const float2 av = *(const float2*)(x + (size_t)row * NBUG + k0 + hk);
        a.x = av.x; a.y = av.y;
        const float2 bv = wB[(size_t)((k0 + hk) >> 1) * NPAD + n];
        b.x = bv.x; b.y = bv.y;
        c = __builtin_amdgcn_wmma_f32_16x16x4_f32(false, a, false, b,
                                                  (short)0, c, false, false);
    }
    const int mbase = bt * 16 + ((lane < 16) ? 0 : 8);
#pragma unroll
    for (int r = 0; r < 8; ++r)
        g[(size_t)(mbase + r) * NPAD + n] = c[r];
}

// ---------------------------------------------------------------------------
// Stage 3: fused NAM. One wave per (batch-tile, k). Loop l in [0,40).
// Both p sub-nets processed together (double-buffered LDS slabs) so two
// independent 4-WMMA chains are in flight and only one s_wait_dscnt per layer
// handoff is needed. DS ops are in-order within a wave, so next-iteration LDS
// writes cannot bypass this iteration's reads (no trailing wait needed).
//   h1 = elu(g*W1+b1)            built per-lane directly in A-layout
//   h2 = elu(h1@W2 + b2)         4x chained wmma_f32_16x16x4, LDS transpose
//   h3 = elu(h2@W3pad + b3)      4x chained wmma (W3 zero-padded 8->16 cols)
//   f  = h3 . W4 + b4            per-lane partial, LDS reduce at the very end
//   out[b,k] = beta[k] + sum_{l,p} sigmoid(alpha[l,k]) * f
// ---------------------------------------------------------------------------
__global__ void nam_kernel(const float* __restrict__ g,
                           const float* __restrict__ alpha,
                           const float* __restrict__ beta,
                           const float* __restrict__ W1, const float* __restrict__ b1,
                           const float* __restrict__ W2, const float* __restrict__ b2,
                           const float* __restrict__ W3, const float* __restrict__ b3,
                           const float* __restrict__ W4, const float* __restrict__ b4,
                           float* __restrict__ out) {
    __shared__ float lds[P_SUB][16 * 17];  // h2 staging per p (C-layout -> A-layout)
    __shared__ float red[32 * 8];          // final cross-lane reduction

    const int bt   = blockIdx.x;     // 0..127
    const int k    = blockIdx.y;     // 0..19
    const int lane = threadIdx.x;    // 0..31
    const int m    = lane & 15;      // A-frag row / C-frag column index
    const int half = lane >> 4;      // 0/1
    const int hk   = half * 2;

    // acc[r]: partial of f_out for output row (bt*16 + r + 8*half), channel m (<8 valid)
    v8f acc = {};

    for (int l = 0; l < L_CL; ++l) {
        const float gv    = g[(size_t)(bt * 16 + m) * NPAD + l];
        const float a_act = sigmoidf_(alpha[l * K_OUT + k]);
        const int   base0 = (k * L_CL + l) * P_SUB;          // (k,l,0) flat index

        // ---- layer 2 for both p: h2 = elu(h1 @ W2 + b2) -> LDS slabs ----
#pragma unroll
        for (int p = 0; p < P_SUB; ++p) {
            const int base = base0 + p;
            const float* w1  = W1 + (size_t)base * 16;
            const float* bb1 = b1 + (size_t)base * 16;
            const float* w2  = W2 + (size_t)base * 256;
            const float* bb2 = b2 + (size_t)base * 16;

            v8f cfrag = {};
#pragma unroll
            for (int c = 0; c < 4; ++c) {
                const int k0 = c * 4 + hk;
                v2f a, b;
                a.x = eluf_(gv * w1[k0 + 0] + bb1[k0 + 0]);
                a.y = eluf_(gv * w1[k0 + 1] + bb1[k0 + 1]);
                b.x = w2[(k0 + 0) * 16 + m];
                b.y = w2[(k0 + 1) * 16 + m];
                cfrag = __builtin_amdgcn_wmma_f32_16x16x4_f32(false, a, false, b,
                                                              (short)0, cfrag, false, false);
            }
            const float bb = bb2[m];
#pragma unroll
            for (int r = 0; r < 8; ++r)
                lds[p][(r + 8 * half) * 17 + m] = eluf_(cfrag[r] + bb);
        }
        asm volatile("s_wait_dscnt 0" ::: "memory");

        // ---- layer 3 for both p: h3 = elu(h2 @ W3pad + b3); head accumulate ----
#pragma unroll
        for (int p = 0; p < P_SUB; ++p) {
            const int base = base0 + p;
            const float* w3  = W3 + (size_t)base * 128;
            const float* bb3 = b3 + (size_t)base * 8;
            const float* w4  = W4 + (size_t)base * 8;

            v8f dfrag = {};
#pragma unroll
            for (int c = 0; c < 4; ++c) {
                const int k0 = c * 4 + hk;
                v2f a, b;
                a.x = lds[p][m * 17 + k0 + 0];
                a.y = lds[p][m * 17 + k0 + 1];
                b.x = (m < 8) ? w3[(k0 + 0) * 8 + m] : 0.0f;
                b.y = (m < 8) ? w3[(k0 + 1) * 8 + m] : 0.0f;
                dfrag = __builtin_amdgcn_wmma_f32_16x16x4_f32(false, a, false, b,
                                                              (short)0, dfrag, false, false);
            }

            if (m < 8) {
                const float b3v = bb3[m];
                const float w4v = w4[m];
#pragma unroll
                for (int r = 0; r < 8; ++r)
                    acc[r] += eluf_(dfrag[r] + b3v) * w4v * a_act;
            }
            if (m == 0) {  // b4 added once per output row (lane0 -> rows 0..7, lane16 -> 8..15)
                const float b4v = b4[base];
#pragma unroll
                for (int r = 0; r < 8; ++r)
                    acc[r] += b4v * a_act;
            }
        }
    }

    // ---- reduce the 16 channel-lanes of each half-wave ----
#pragma unroll
    for (int r = 0; r < 8; ++r)
        red[lane * 8 + r] = acc[r];
    asm volatile("s_wait_dscnt 0" ::: "memory");

    if (lane < 16) {
        const int mr = lane;                 // output row within tile
        const int lo = (mr < 8) ? 0 : 16;    // lanes holding this row group
        const int r  = mr & 7;
        float s = 0.0f;
#pragma unroll
        for (int j = 0; j < 16; ++j)
            s += red[(lo + j) * 8 + r];
        out[(size_t)(bt * 16 + mr) * K_OUT + k] = beta[k] + s;
    }
}

// ---------------------------------------------------------------------------
extern "C" void kernel_launch(void* const* d_in, const int* in_sizes, int n_in,
                              void* d_out, int out_size, void* d_ws, size_t ws_size,
                              hipStream_t stream) {
    const float* x     = (const float*)d_in[0];
    const float* ml    = (const float*)d_in[1];
    const float* mu    = (const float*)d_in[2];
    const float* rbug  = (const float*)d_in[3];
    const float* alpha = (const float*)d_in[4];
    const float* beta  = (const float*)d_in[5];
    const float* W1    = (const float*)d_in[6];
    const float* b1    = (const float*)d_in[7];
    const float* W2    = (const float*)d_in[8];
    const float* b2    = (const float*)d_in[9];
    const float* W3    = (const float*)d_in[10];
    const float* b3    = (const float*)d_in[11];
    const float* W4    = (const float*)d_in[12];
    const float* b4    = (const float*)d_in[13];
    float* out = (float*)d_out;

    // Workspace: wactB [1000][48][2] then g [2048][48]  (~780 KB total)
    float* wactB = (float*)d_ws;
    float* g     = wactB + (size_t)NBUG * NPAD;

    // Stage 1
    {
        int total = NBUG * NPAD;
        wact_kernel<<<(total + 255) / 256, 256, 0, stream>>>(ml, mu, rbug, wactB);
    }
    // Stage 2: g = x @ w_act  (WMMA f32 16x16x4)
    {
        dim3 grid(B_SZ / 16, NPAD / 16);  // 128 x 3
        g_gemm_kernel<<<grid, 32, 0, stream>>>(x, wactB, g);
    }
    // Stage 3: fused NAM (WMMA f32 16x16x4) -> out [2048][20]
    {
        dim3 grid(B_SZ / 16, K_OUT);      // 128 x 20
        nam_kernel<<<grid, 32, 0, stream>>>(g, alpha, beta,
                                            W1, b1, W2, b2, W3, b3, W4, b4, out);
    }
}